// GradientOutput_76012331204783
// MI455X (gfx1250) — compile-verified
//
#include <hip/hip_runtime.h>

// GradientOutput (forces + virial stress) for MI455X / gfx1250.
//
// Memory-bound scatter (AI ~ 1 FLOP/byte, ~105 MB streamed, ~4.5us floor at
// 23.3 TB/s). Strategy:
//  - streaming inputs (edge_diff, dE_ddiff, edge_idx) pushed memory->LDS with
//    gfx1250 async-to-LDS ops (ASYNCcnt), double-buffered one tile ahead.
//  - forces: global_atomic_add_f32, 6 per edge, ~300K distinct lines in the
//    192MB L2.
//  - stress: per-wave private 16x9 LDS accumulators with ds_add_f32, one
//    global flush per block. Same-image edges cancel exactly -> skipped.

constexpr int TPB   = 256;         // 8 wave32 waves per block
constexpr int NWAVE = TPB / 32;
constexpr int BCAP  = 16;          // images held in LDS (B == 16 here)

#if defined(__gfx1250__) && defined(__has_builtin)
#if __has_builtin(__builtin_amdgcn_global_load_async_to_lds_b32) && \
    __has_builtin(__builtin_amdgcn_global_load_async_to_lds_b64)
#define USE_ASYNC_LDS 1
#endif
#endif

#ifdef USE_ASYNC_LDS
#define AS1 __attribute__((address_space(1)))
#define AS3 __attribute__((address_space(3)))
typedef int v2i __attribute__((ext_vector_type(2)));

__device__ __forceinline__ void async_b32(const float* g, float* l) {
  // signature (from clang diagnostics): (AS1 int*, AS3 int*, Ii offset, Ii cpol)
  __builtin_amdgcn_global_load_async_to_lds_b32(
      (AS1 int*)g, (AS3 int*)l, 0, 0);
}
__device__ __forceinline__ void async_b64(const int* g, int* l) {
  // signature: (AS1 v2i*, AS3 v2i*, Ii offset, Ii cpol)
  __builtin_amdgcn_global_load_async_to_lds_b64(
      (AS1 v2i*)g, (AS3 v2i*)l, 0, 0);
}
template <int N>
__device__ __forceinline__ void wait_asynccnt() {
#if __has_builtin(__builtin_amdgcn_s_wait_asynccnt)
  __builtin_amdgcn_s_wait_asynccnt(N);
#else
  asm volatile("s_wait_asynccnt %0" ::"i"(N) : "memory");
#endif
}
#endif  // USE_ASYNC_LDS

__global__ void zero_f32(float* __restrict__ a, int n,
                         float* __restrict__ b, int m) {
  int i = blockIdx.x * blockDim.x + threadIdx.x;
  if (i < n) a[i] = 0.0f;
  if (i < m) b[i] = 0.0f;
}

__global__ __launch_bounds__(TPB)
void edge_scatter(const float* __restrict__ edge_diff,  // [E,3]
                  const float* __restrict__ dE,         // [E,3]
                  const int*   __restrict__ eidx,       // [E,2]
                  const int*   __restrict__ image_idx,  // [N]
                  float*       __restrict__ forces,     // [N,3], pre-zeroed
                  float*       __restrict__ stress_ws,  // [B,3,3], pre-zeroed
                  int E, int B) {
  __shared__ float s_stress[NWAVE][BCAP * 9];
  const int tid  = threadIdx.x;
  const int wave = tid >> 5;

  for (int k = tid; k < NWAVE * BCAP * 9; k += TPB)
    (&s_stress[0][0])[k] = 0.0f;

  // --------- per-edge processing (forces + per-wave stress tiles) ---------
  auto process = [&](int i, int j, float dx, float dy, float dz,
                     float gx, float gy, float gz) {
    atomicAdd(&forces[3 * i + 0],  gx);
    atomicAdd(&forces[3 * i + 1],  gy);
    atomicAdd(&forces[3 * i + 2],  gz);
    atomicAdd(&forces[3 * j + 0], -gx);
    atomicAdd(&forces[3 * j + 1], -gy);
    atomicAdd(&forces[3 * j + 2], -gz);

    const int bi = image_idx[i];
    const int bj = image_idx[j];
    if (bi != bj) {  // same-image contributions cancel exactly
      float o[9] = { dx * gx, dx * gy, dx * gz,
                     dy * gx, dy * gy, dy * gz,
                     dz * gx, dz * gy, dz * gz };
      if ((unsigned)bi < (unsigned)BCAP && (unsigned)bj < (unsigned)BCAP) {
        float* si = &s_stress[wave][bi * 9];
        float* sj = &s_stress[wave][bj * 9];
#pragma unroll
        for (int k = 0; k < 9; ++k) {       // ds_add_f32, per-wave copy
          atomicAdd(&sj[k],  o[k]);
          atomicAdd(&si[k], -o[k]);
        }
      } else {  // safety fallback if B ever exceeds BCAP
#pragma unroll
        for (int k = 0; k < 9; ++k) {
          atomicAdd(&stress_ws[bj * 9 + k],  o[k]);
          atomicAdd(&stress_ws[bi * 9 + k], -o[k]);
        }
      }
    }
  };

#ifdef USE_ASYNC_LDS
  // Double-buffered async memory->LDS staging of the edge stream.
  // Per tile (TPB edges): idx 2KB + diff 3KB + grad 3KB, 7 async ops/wave.
  __shared__ int   l_idx[2][TPB * 2];
  __shared__ float l_d[2][TPB * 3];
  __shared__ float l_g[2][TPB * 3];

  const long tiles = ((long)E + TPB - 1) / TPB;
  const long e3max = 3L * E - 1;

  auto issue = [&](long tile, int buf) {
    long e   = tile * TPB + tid;
    long dwp = (e < E) ? 2 * e : 0;             // clamp tail (never OOB)
    async_b64(eidx + dwp, &l_idx[buf][2 * tid]);
    const long base3 = tile * (TPB * 3);
#pragma unroll
    for (int k = 0; k < 3; ++k) {
      long dw = base3 + tid + k * TPB;
      if (dw > e3max) dw = 0;
      async_b32(edge_diff + dw, &l_d[buf][tid + k * TPB]);
      async_b32(dE        + dw, &l_g[buf][tid + k * TPB]);
    }
  };

  __syncthreads();  // s_stress zero + LDS buffers ready
  long t0 = blockIdx.x;
  if (t0 < tiles) issue(t0, 0);

  int buf = 0;
  for (long tile = t0; tile < tiles; tile += gridDim.x) {
    const long next = tile + gridDim.x;
    if (next < tiles) {
      issue(next, buf ^ 1);       // 7 more in flight
      wait_asynccnt<7>();         // this wave's tile-n ops complete
    } else {
      wait_asynccnt<0>();
    }
    __syncthreads();              // all waves' tile-n data visible in LDS

    const long e = tile * TPB + tid;
    if (e < E) {
      const int   i  = l_idx[buf][2 * tid + 0];
      const int   j  = l_idx[buf][2 * tid + 1];
      const float dx = l_d[buf][3 * tid + 0];
      const float dy = l_d[buf][3 * tid + 1];
      const float dz = l_d[buf][3 * tid + 2];
      const float gx = l_g[buf][3 * tid + 0];
      const float gy = l_g[buf][3 * tid + 1];
      const float gz = l_g[buf][3 * tid + 2];
      process(i, j, dx, dy, dz, gx, gy, gz);
    }
    __syncthreads();              // buffer may be re-filled next iteration
    buf ^= 1;
  }
#else
  // Fallback: direct nontemporal streaming loads + prefetch.
  __syncthreads();
  const int stride = gridDim.x * TPB;
  for (int e = blockIdx.x * TPB + tid; e < E; e += stride) {
    int en = e + stride;
    if (en < E) {
      __builtin_prefetch(edge_diff + 3 * en, 0, 1);
      __builtin_prefetch(dE + 3 * en, 0, 1);
      __builtin_prefetch(eidx + 2 * en, 0, 1);
    }
    const int   i  = __builtin_nontemporal_load(&eidx[2 * e + 0]);
    const int   j  = __builtin_nontemporal_load(&eidx[2 * e + 1]);
    const float dx = __builtin_nontemporal_load(&edge_diff[3 * e + 0]);
    const float dy = __builtin_nontemporal_load(&edge_diff[3 * e + 1]);
    const float dz = __builtin_nontemporal_load(&edge_diff[3 * e + 2]);
    const float gx = __builtin_nontemporal_load(&dE[3 * e + 0]);
    const float gy = __builtin_nontemporal_load(&dE[3 * e + 1]);
    const float gz = __builtin_nontemporal_load(&dE[3 * e + 2]);
    process(i, j, dx, dy, dz, gx, gy, gz);
  }
#endif

  __syncthreads();
  // cross-wave reduce + one global flush per block (B*9 atomics)
  const int lim = (B < BCAP ? B : BCAP) * 9;
  for (int k = tid; k < lim; k += TPB) {
    float acc = 0.0f;
#pragma unroll
    for (int w = 0; w < NWAVE; ++w) acc += s_stress[w][k];
    atomicAdd(&stress_ws[k], acc);
  }
}

__global__ void finalize_voigt(const float* __restrict__ cell,   // [B,3,3]
                               const float* __restrict__ sws,    // [B,3,3]
                               float* __restrict__ out_voigt,    // [B,6]
                               int B) {
  int t = blockIdx.x * blockDim.x + threadIdx.x;
  if (t >= B * 6) return;
  const int b = t / 6, v = t % 6;
  const float* c = cell + 9 * b;
  // volume = row0 . cross(row1, row2)
  const float cr0 = c[4] * c[8] - c[5] * c[7];
  const float cr1 = c[5] * c[6] - c[3] * c[8];
  const float cr2 = c[3] * c[7] - c[4] * c[6];
  const float vol = c[0] * cr0 + c[1] * cr1 + c[2] * cr2;
  const float* s = sws + 9 * b;
  float val;
  switch (v) {
    case 0:  val = s[0]; break;                      // xx
    case 1:  val = s[4]; break;                      // yy
    case 2:  val = s[8]; break;                      // zz
    case 3:  val = 0.5f * (s[5] + s[7]); break;      // yz
    case 4:  val = 0.5f * (s[2] + s[6]); break;      // xz
    default: val = 0.5f * (s[1] + s[3]); break;      // xy
  }
  out_voigt[t] = val / (vol * 2.0f);
}

extern "C" void kernel_launch(void* const* d_in, const int* in_sizes, int n_in,
                              void* d_out, int out_size, void* d_ws, size_t ws_size,
                              hipStream_t stream) {
  const float* edge_diff = (const float*)d_in[0];   // [E,3] f32
  const float* dE        = (const float*)d_in[1];   // [E,3] f32
  const float* cell      = (const float*)d_in[2];   // [B,3,3] f32
  const int*   eidx      = (const int*)  d_in[3];   // [E,2] i32
  const int*   image_idx = (const int*)  d_in[4];   // [N] i32
  // d_in[5] = n_atoms scalar (N recovered from in_sizes[4])

  const int E = in_sizes[0] / 3;
  const int B = in_sizes[2] / 9;
  const int N = in_sizes[4];

  float* forces    = (float*)d_out;                       // [N,3]
  float* out_voigt = (float*)d_out + (size_t)N * 3;       // [B,6]
  float* stress_ws = (float*)d_ws;                        // [B,3,3]

  // deterministic re-init every launch (graph replay safe)
  const int zm   = B * 9;
  const int zmax = out_size > zm ? out_size : zm;
  zero_f32<<<(zmax + TPB - 1) / TPB, TPB, 0, stream>>>(
      (float*)d_out, out_size, stress_ws, zm);

  int nblocks = (E + TPB - 1) / TPB;
  if (nblocks > 4096) nblocks = 4096;   // ~3 tiles per block, double-buffered
  edge_scatter<<<nblocks, TPB, 0, stream>>>(
      edge_diff, dE, eidx, image_idx, forces, stress_ws, E, B);

  finalize_voigt<<<(B * 6 + 127) / 128, 128, 0, stream>>>(
      cell, stress_ws, out_voigt, B);
}